// SeriesEmbedding_71227737637031
// MI455X (gfx1250) — compile-verified
//
#include <hip/hip_runtime.h>

typedef __attribute__((ext_vector_type(2))) float v2f;
typedef __attribute__((ext_vector_type(4))) float v4f;
typedef __attribute__((ext_vector_type(8))) float v8f;

namespace {
constexpr int kB = 128, kL = 336, kN = 321, kE = 128, kLAT = 64;
constexpr int kKC  = 48;        // stage-1 K chunk (divides 336, multiple of 4)
constexpr int kJ1  = kKC / 4;   // 12 k-steps per chunk
constexpr int kWLD = kKC + 2;   // 50: padded LDS stride for W_embed chunk [e][k]
constexpr int kHLD = kE + 2;    // 130: padded LDS stride for h_hat [b][e] and W_mu/var [lat][e]
constexpr int kThreads = 256;   // 8 wave32
constexpr size_t kSmemFloats = (size_t)kB * kHLD + 2ul * kLAT * kHLD; // 16640 + 16640
constexpr size_t kXpFloats   = (size_t)kN * kB * kL;                  // transposed x scratch
}

// ---------------------------------------------------------------------------
// Pre-pass: xp[n][b][l] = x[b][l][n]
// ---------------------------------------------------------------------------
__global__ __launch_bounds__(256) void transpose_x_kernel(
    const float* __restrict__ x,   // [B, L, N]
    float* __restrict__ xp)        // [N, B, L]
{
  __shared__ float tile[32][33];
  const int n0 = blockIdx.x * 32;
  const int l0 = blockIdx.y * 32;
  const int b  = blockIdx.z;
  const int tx = threadIdx.x & 31;
  const int ty = threadIdx.x >> 5;       // 0..7

#pragma unroll
  for (int j = 0; j < 4; ++j) {          // read coalesced along n
    const int ll = l0 + ty + 8 * j;
    const int nn = n0 + tx;
    float v = 0.f;
    if (ll < kL && nn < kN) v = x[((size_t)b * kL + ll) * kN + nn];
    tile[ty + 8 * j][tx] = v;
  }
  __syncthreads();
#pragma unroll
  for (int j = 0; j < 4; ++j) {          // write coalesced along l
    const int r  = ty + 8 * j;
    const int nn = n0 + r;
    const int ll = l0 + tx;
    if (nn < kN && ll < kL)
      xp[((size_t)nn * kB + b) * kL + ll] = tile[tx][r];
  }
}

// ---------------------------------------------------------------------------
// Fused grouped-GEMM kernel. Wave mapping: 8 waves = 4 row-slabs (32 rows)
// x 2 column halves -> each B fragment feeds 2 WMMAs (halves LDS traffic).
// XT=true: A from xp [N,B,L] (contiguous b64). XT=false: gather fallback.
// ---------------------------------------------------------------------------
template <bool XT>
__global__ __launch_bounds__(kThreads) void series_embed_fused(
    const float* __restrict__ xsrc,     // xp [N,B,L] if XT else x [B,L,N]
    const float* __restrict__ time_x,   // [B, N, E]
    const float* __restrict__ W_embed,  // [N, L, E]
    const float* __restrict__ b_embed,  // [N, E]
    const float* __restrict__ W_mu,     // [N, E, LAT]
    const float* __restrict__ b_mu,     // [N, LAT]
    const float* __restrict__ W_var,    // [N, E, LAT]
    const float* __restrict__ b_var,    // [N, LAT]
    float* __restrict__ out_h,          // [B, N, E]
    float* __restrict__ out_hh,         // [B, N, E]
    float* __restrict__ out_mu,         // [B, N, LAT]
    float* __restrict__ out_var)        // [B, N, LAT]
{
  extern __shared__ float smem[];
  float* hh_lds = smem;                       // [B][kHLD]  (h_hat staging for stage 2)
  float* uni    = smem + (size_t)kB * kHLD;   // aliased region
  float* w_lds  = uni;                        // stage 1: [E][kWLD] transposed W_embed chunk
  float* wm_lds = uni;                        // stage 2: [LAT][kHLD] transposed W_mu
  float* wv_lds = uni + (size_t)kLAT * kHLD;  // stage 2: [LAT][kHLD] transposed W_var

  const int n    = blockIdx.x;                // channel
  const int tid  = threadIdx.x;
  const int wave = tid >> 5;                  // 0..7
  const int lane = tid & 31;
  const int lg   = lane >> 4;                 // lane group (0/1)
  const int lm   = lane & 15;

  const int row0 = (wave & 3) * 32;           // 32-row slab (2 sub-tiles of 16)
  const int ch   = wave >> 2;                 // column half (0/1)
  const int c1   = ch * 64;                   // stage-1 E-column base (4 tiles)
  const int c2   = ch * 32;                   // stage-2 LAT-column base (2 tiles)

  const v8f vzero = {0.f, 0.f, 0.f, 0.f, 0.f, 0.f, 0.f, 0.f};

  // Per-lane A row bases (two 16-row sub-tiles)
  const float* arow0 = XT ? (xsrc + ((size_t)n * kB + row0 + lm) * kL)
                          : (xsrc + ((size_t)(row0 + lm) * kL) * kN + n);
  const float* arow1 = XT ? (arow0 + (size_t)16 * kL)
                          : (arow0 + (size_t)16 * kL * kN);

  auto loadA1 = [&](const float* base, int l) -> v2f {  // A frag: row lm, k=2*lg+{0,1}
    if (XT) {
      return *(const v2f*)&base[l + 2 * lg];
    } else {
      const float* ap = base + (size_t)(l + 2 * lg) * kN;
      v2f a; a.x = ap[0]; a.y = ap[kN];
      return a;
    }
  };

  // -------- Stage 1: h[b,e] = sum_l x[b,l,n] * W_embed[n,l,e] --------
  v8f acc[2][4];
#pragma unroll
  for (int s = 0; s < 2; ++s)
#pragma unroll
    for (int t = 0; t < 4; ++t) acc[s][t] = vzero;

  for (int l0 = 0; l0 < kL; l0 += kKC) {
    // Cooperative load of W_embed[n, l0:l0+KC, :] into LDS, transposed [e][k].
    for (int i = tid; i < kKC * (kE / 4); i += kThreads) {
      const int kk = i >> 5;            // 0..47
      const int e4 = (i & 31) << 2;     // 0,4,...,124
      const v4f w = *(const v4f*)&W_embed[((size_t)n * kL + (l0 + kk)) * kE + e4];
      w_lds[(e4 + 0) * kWLD + kk] = w.x;
      w_lds[(e4 + 1) * kWLD + kk] = w.y;
      w_lds[(e4 + 2) * kWLD + kk] = w.z;
      w_lds[(e4 + 3) * kWLD + kk] = w.w;
    }
    __syncthreads();

#pragma unroll
    for (int j = 0; j < kJ1; ++j) {              // 12 k-steps of 4
      const int l = l0 + 4 * j;
      const v2f aLo = loadA1(arow0, l);
      const v2f aHi = loadA1(arow1, l);
#pragma unroll
      for (int t = 0; t < 4; ++t) {              // 4 E-column tiles in this half
        const v2f b = *(const v2f*)&w_lds[(c1 + 16 * t + lm) * kWLD + 4 * j + 2 * lg];
        acc[0][t] = __builtin_amdgcn_wmma_f32_16x16x4_f32(
            false, aLo, false, b, (short)0, acc[0][t], false, false);
        acc[1][t] = __builtin_amdgcn_wmma_f32_16x16x4_f32(
            false, aHi, false, b, (short)0, acc[1][t], false, false);
      }
    }
    __syncthreads();  // protect w_lds before next chunk overwrite
  }

  // -------- Epilogue 1: bias, sigmoid gate, write h / h_hat, stage h_hat --------
  // C/D layout: VGPR r -> row r + 8*lg, col = lm (per 16-col tile)
#pragma unroll
  for (int s = 0; s < 2; ++s) {
#pragma unroll
    for (int t = 0; t < 4; ++t) {
      const int e = c1 + 16 * t + lm;
      const float be = b_embed[(size_t)n * kE + e];
#pragma unroll
      for (int r = 0; r < 8; ++r) {
        const int brow = row0 + 16 * s + r + 8 * lg;
        const size_t oidx = ((size_t)brow * kN + n) * kE + e;
        const float hv = acc[s][t][r] + be;
        const float tx = time_x[oidx];
        const float sg = 1.0f / (1.0f + __expf(-tx));
        const float hh = sg * hv;
        out_h[oidx]  = hv;
        out_hh[oidx] = hh;
        hh_lds[(size_t)brow * kHLD + e] = hh;
      }
    }
  }

  // Cooperative load of W_mu / W_var, transposed into [lat][e] (stride kHLD).
  for (int i = tid; i < kE * (kLAT / 4); i += kThreads) {
    const int e  = i >> 4;              // 0..127
    const int l4 = (i & 15) << 2;       // 0,4,...,60
    const v4f m = *(const v4f*)&W_mu [((size_t)n * kE + e) * kLAT + l4];
    const v4f v = *(const v4f*)&W_var[((size_t)n * kE + e) * kLAT + l4];
    wm_lds[(l4 + 0) * kHLD + e] = m.x;
    wm_lds[(l4 + 1) * kHLD + e] = m.y;
    wm_lds[(l4 + 2) * kHLD + e] = m.z;
    wm_lds[(l4 + 3) * kHLD + e] = m.w;
    wv_lds[(l4 + 0) * kHLD + e] = v.x;
    wv_lds[(l4 + 1) * kHLD + e] = v.y;
    wv_lds[(l4 + 2) * kHLD + e] = v.z;
    wv_lds[(l4 + 3) * kHLD + e] = v.w;
  }
  __syncthreads();

  // -------- Stage 2: mu/var[b,k] = sum_e h_hat[b,e] * W_{mu,var}[n,e,k] --------
  v8f amu[2][2], avr[2][2];
#pragma unroll
  for (int s = 0; s < 2; ++s)
#pragma unroll
    for (int t = 0; t < 2; ++t) { amu[s][t] = vzero; avr[s][t] = vzero; }

#pragma unroll
  for (int j = 0; j < kE / 4; ++j) {             // 32 k-steps of 4 over e
    const v2f aLo = *(const v2f*)&hh_lds[(size_t)(row0 + lm)      * kHLD + 4 * j + 2 * lg];
    const v2f aHi = *(const v2f*)&hh_lds[(size_t)(row0 + 16 + lm) * kHLD + 4 * j + 2 * lg];
#pragma unroll
    for (int t = 0; t < 2; ++t) {                // 2 LAT-column tiles in this half
      const v2f bm = *(const v2f*)&wm_lds[(c2 + 16 * t + lm) * kHLD + 4 * j + 2 * lg];
      const v2f bv = *(const v2f*)&wv_lds[(c2 + 16 * t + lm) * kHLD + 4 * j + 2 * lg];
      amu[0][t] = __builtin_amdgcn_wmma_f32_16x16x4_f32(
          false, aLo, false, bm, (short)0, amu[0][t], false, false);
      amu[1][t] = __builtin_amdgcn_wmma_f32_16x16x4_f32(
          false, aHi, false, bm, (short)0, amu[1][t], false, false);
      avr[0][t] = __builtin_amdgcn_wmma_f32_16x16x4_f32(
          false, aLo, false, bv, (short)0, avr[0][t], false, false);
      avr[1][t] = __builtin_amdgcn_wmma_f32_16x16x4_f32(
          false, aHi, false, bv, (short)0, avr[1][t], false, false);
    }
  }

  // -------- Epilogue 2: bias, write mu / var --------
#pragma unroll
  for (int s = 0; s < 2; ++s) {
#pragma unroll
    for (int t = 0; t < 2; ++t) {
      const int lat = c2 + 16 * t + lm;
      const float bm = b_mu [(size_t)n * kLAT + lat];
      const float bv = b_var[(size_t)n * kLAT + lat];
#pragma unroll
      for (int r = 0; r < 8; ++r) {
        const int brow = row0 + 16 * s + r + 8 * lg;
        const size_t oidx = ((size_t)brow * kN + n) * kLAT + lat;
        out_mu[oidx]  = amu[s][t][r] + bm;
        out_var[oidx] = avr[s][t][r] + bv;
      }
    }
  }
}

extern "C" void kernel_launch(void* const* d_in, const int* in_sizes, int n_in,
                              void* d_out, int out_size, void* d_ws, size_t ws_size,
                              hipStream_t stream) {
  (void)in_sizes; (void)n_in; (void)out_size;

  const float* x       = (const float*)d_in[0];
  const float* time_x  = (const float*)d_in[1];
  const float* W_embed = (const float*)d_in[2];
  const float* b_embed = (const float*)d_in[3];
  const float* W_mu    = (const float*)d_in[4];
  const float* b_mu    = (const float*)d_in[5];
  const float* W_var   = (const float*)d_in[6];
  const float* b_var   = (const float*)d_in[7];

  float* out = (float*)d_out;
  float* out_h   = out;
  float* out_hh  = out_h  + (size_t)kB * kN * kE;
  float* out_mu  = out_hh + (size_t)kB * kN * kE;
  float* out_var = out_mu + (size_t)kB * kN * kLAT;

  const size_t smem_bytes = kSmemFloats * sizeof(float);  // 133120 B < 320 KB/WGP
  const size_t xp_bytes   = kXpFloats * sizeof(float);    // ~55.3 MB scratch

  if (ws_size >= xp_bytes) {
    float* xp = (float*)d_ws;
    hipLaunchKernelGGL(transpose_x_kernel,
                       dim3((kN + 31) / 32, (kL + 31) / 32, kB), dim3(256), 0, stream,
                       x, xp);
    hipLaunchKernelGGL((series_embed_fused<true>), dim3(kN), dim3(kThreads), smem_bytes,
                       stream, (const float*)xp, time_x, W_embed, b_embed,
                       W_mu, b_mu, W_var, b_var, out_h, out_hh, out_mu, out_var);
  } else {
    hipLaunchKernelGGL((series_embed_fused<false>), dim3(kN), dim3(kThreads), smem_bytes,
                       stream, x, time_x, W_embed, b_embed,
                       W_mu, b_mu, W_var, b_var, out_h, out_hh, out_mu, out_var);
  }
}